// HGCLD_15788299780622
// MI455X (gfx1250) — compile-verified
//
#include <hip/hip_runtime.h>
#include <math.h>

typedef __attribute__((ext_vector_type(2))) float v2f;
typedef __attribute__((ext_vector_type(8))) float v8f;

#define N_U_NODES 50000
#define N_V_NODES 50000
#define N_NODES   (N_U_NODES + N_V_NODES)   // 100000
#define D         128
#define ROW_TILES_U   (N_U_NODES / 16)      // 3125 (exact)
#define ROW_TILES_TOT (N_NODES / 16)        // 6250 (exact)

// ---------------- degree accumulation (deg_out == deg_in by construction) ----
__global__ void HGCLD_deg_kernel(const int* __restrict__ src,
                                 float* __restrict__ deg, int E) {
    int e = blockIdx.x * blockDim.x + threadIdx.x;
    if (e < E) atomicAdd(&deg[src[e]], 1.0f);
}

// ---------------- projection GEMM: node_f = X @ W, scaled by deg^-0.5 -------
// One block = 8 waves; block b handles rows [16b, 16b+16), wave w handles
// output columns [16w, 16w+16). K-loop: 32 x v_wmma_f32_16x16x4_f32.
__global__ void __launch_bounds__(256)
HGCLD_proj_kernel(const float* __restrict__ u_f, const float* __restrict__ v_f,
                  const float* __restrict__ u_w, const float* __restrict__ v_w,
                  const float* __restrict__ deg, float* __restrict__ node_f) {
    const int wave = threadIdx.x >> 5;   // 0..7 -> column tile
    const int lane = threadIdx.x & 31;
    const int half = lane >> 4;          // 0: K=k0..k0+1 / M=r ; 1: K=k0+2..k0+3 / M=r+8
    const int lr   = lane & 15;

    const int rowTile = blockIdx.x;      // 0..6249 (block-uniform branch below)
    const bool isU = rowTile < ROW_TILES_U;
    const float* __restrict__ X = isU ? u_f : v_f;
    const float* __restrict__ W = isU ? u_w : v_w;
    const int localRowBase  = (isU ? rowTile : (rowTile - ROW_TILES_U)) * 16;
    const int globalRowBase = rowTile * 16;
    const int colBase = wave * 16;

    // A-matrix 16x4 f32 layout: lane lr holds row M=lr; VGPR0/1 = K pair.
    const float* __restrict__ arow = X + (size_t)(localRowBase + lr) * D;

    v8f c = {};
    #pragma unroll
    for (int k0 = 0; k0 < D; k0 += 4) {
        const int ka = k0 + 2 * half;
        v2f a = *(const v2f*)(arow + ka);               // {A[M][ka], A[M][ka+1]}
        v2f b;
        b.x = W[(size_t)ka * D + colBase + lr];         // B[ka][n]
        b.y = W[(size_t)(ka + 1) * D + colBase + lr];   // B[ka+1][n]
        // 8 args: (neg_a, A, neg_b, B, c_mod, C, reuse_a, reuse_b)
        c = __builtin_amdgcn_wmma_f32_16x16x4_f32(false, a, false, b,
                                                  (short)0, c, false, false);
    }

    // C/D layout: VGPR r, lanes 0-15 -> M=r, lanes 16-31 -> M=r+8; N = lr.
    #pragma unroll
    for (int r = 0; r < 8; ++r) {
        const int row = globalRowBase + r + 8 * half;
        const float dg = deg[row];
        const float s = rsqrtf(dg < 1.0f ? 1.0f : dg);
        node_f[(size_t)row * D + colBase + lr] = c[r] * s;
    }
}

// ---------------- edge scatter: out[dst] += node_f[src] ---------------------
// One wave per edge: 32 lanes x float4 = 512B row. node_f + out fit in L2.
__global__ void HGCLD_scatter_kernel(const float* __restrict__ node_f,
                                     const int* __restrict__ src,
                                     const int* __restrict__ dst,
                                     float* __restrict__ out, int E) {
    const long long gid = (long long)blockIdx.x * blockDim.x + threadIdx.x;
    const int e    = (int)(gid >> 5);
    const int lane = (int)(gid & 31);
    if (e >= E) return;
    const int s = src[e];
    const int d = dst[e];
    const float4 v = *(const float4*)(node_f + (size_t)s * D + lane * 4);
    float* o = out + (size_t)d * D + lane * 4;
    atomicAdd(o + 0, v.x);
    atomicAdd(o + 1, v.y);
    atomicAdd(o + 2, v.z);
    atomicAdd(o + 3, v.w);
}

// ---------------- final normalization: out *= deg_in^-0.5 -------------------
__global__ void HGCLD_scale_kernel(float* __restrict__ out,
                                   const float* __restrict__ deg) {
    const long long i = (long long)blockIdx.x * blockDim.x + threadIdx.x;
    if (i >= (long long)N_NODES * D) return;
    const int row = (int)(i >> 7);      // D == 128
    const float dg = deg[row];
    out[i] *= rsqrtf(dg < 1.0f ? 1.0f : dg);
}

extern "C" void kernel_launch(void* const* d_in, const int* in_sizes, int n_in,
                              void* d_out, int out_size, void* d_ws, size_t ws_size,
                              hipStream_t stream) {
    const float* u_f = (const float*)d_in[0];
    const float* v_f = (const float*)d_in[1];
    const float* u_w = (const float*)d_in[2];
    const float* v_w = (const float*)d_in[3];
    const int*   src = (const int*)d_in[4];
    const int*   dst = (const int*)d_in[5];
    float* out = (float*)d_out;
    const int E = in_sizes[4];           // 1,600,000 directed edges

    // workspace layout: [deg: N floats][pad to 256B][node_f: N*D floats]
    float* deg = (float*)d_ws;
    size_t deg_bytes = (((size_t)N_NODES * sizeof(float)) + 255) & ~(size_t)255;
    float* node_f = (float*)((char*)d_ws + deg_bytes);

    hipMemsetAsync(deg, 0, (size_t)N_NODES * sizeof(float), stream);
    hipMemsetAsync(out, 0, (size_t)N_NODES * D * sizeof(float), stream);

    HGCLD_deg_kernel<<<(E + 255) / 256, 256, 0, stream>>>(src, deg, E);

    HGCLD_proj_kernel<<<ROW_TILES_TOT, 256, 0, stream>>>(u_f, v_f, u_w, v_w,
                                                         deg, node_f);

    const long long scat_threads = (long long)E * 32;
    HGCLD_scatter_kernel<<<(unsigned)((scat_threads + 255) / 256), 256, 0, stream>>>(
        node_f, src, dst, out, E);

    const long long scale_threads = (long long)N_NODES * D;
    HGCLD_scale_kernel<<<(unsigned)((scale_threads + 255) / 256), 256, 0, stream>>>(
        out, deg);
}